// SummaryStatsRegressor_34376918238066
// MI455X (gfx1250) — compile-verified
//
#include <hip/hip_runtime.h>
#include <stdint.h>

#define NS   16384      // S
#define NF   16         // F
#define NBAT 256        // B
#define NT   1024       // threads per stats block
#define DPOOL 113       // 7*F+1
#define PSTR 116        // pooled row stride (113 padded to multiple of 4)

// ---- dynamic LDS layout for stats kernel ----
#define OFF_KEYS 0          // 16384 * 4 = 65536  (column values, later radix keys)
#define OFF_MK   65536      // 16384 * 1 = 16384  (mask bytes)
#define OFF_HIST 81920      // 2048 * 4 = 8192
#define OFF_SCAN 90112      // 2048 * 4 = 8192
#define OFF_CTRL 98304      // 64 bytes
#define OFF_RED  98368      // 32 * 4 = 128
#define SMEMB    98496

typedef __attribute__((ext_vector_type(2))) float v2f;
typedef __attribute__((ext_vector_type(8))) float v8f;

// Flat address of an LDS object: low 32 bits are the LDS byte offset.
__device__ __forceinline__ unsigned lds_off32(const void* p) {
  return (unsigned)(unsigned long long)p;
}

// CDNA5 async global->LDS copies (ASYNCcnt-tracked), GV addressing mode.
__device__ __forceinline__ void async_ld_b32(unsigned ldsoff, const void* g) {
  asm volatile("global_load_async_to_lds_b32 %0, %1, off"
               :: "v"(ldsoff), "v"(g) : "memory");
}
__device__ __forceinline__ void async_ld_b128(unsigned ldsoff, const void* g) {
  asm volatile("global_load_async_to_lds_b128 %0, %1, off"
               :: "v"(ldsoff), "v"(g) : "memory");
}
__device__ __forceinline__ void wait_async0() {
  asm volatile("s_wait_asynccnt 0x0" ::: "memory");
}

// wave32 + block reduction. op: 0=add 1=min 2=max
__device__ __forceinline__ float red_op(float v, float* buf, int tid, int op) {
  #pragma unroll
  for (int off = 16; off > 0; off >>= 1) {
    float o = __shfl_down(v, off, 32);
    v = (op == 0) ? (v + o) : (op == 1 ? fminf(v, o) : fmaxf(v, o));
  }
  if ((tid & 31) == 0) buf[tid >> 5] = v;
  __syncthreads();
  if (tid < 32) {
    float x = buf[tid];
    #pragma unroll
    for (int off = 16; off > 0; off >>= 1) {
      float o = __shfl_down(x, off, 32);
      x = (op == 0) ? (x + o) : (op == 1 ? fminf(x, o) : fmaxf(x, o));
    }
    if (tid == 0) buf[0] = x;
  }
  __syncthreads();
  float r = buf[0];
  __syncthreads();
  return r;
}

__device__ __forceinline__ float key_to_float(unsigned k) {
  unsigned u = (k & 0x80000000u) ? (k & 0x7FFFFFFFu) : ~k;
  return __uint_as_float(u);
}

// Exact rank-r selection over keys[0..NS) via 3-level radix histograms (11/11/10 bits).
// Collective: all NT threads must call with identical r.
__device__ unsigned select_key(unsigned r, const unsigned* keys, unsigned* hist,
                               unsigned* scanb, unsigned* ctrl, int tid) {
  unsigned prefix = 0;
  int prefix_bits = 0;
  const int shifts[3] = {21, 10, 0};
  const int bitsl[3]  = {11, 11, 10};
  for (int lev = 0; lev < 3; ++lev) {
    const int nb = 1 << bitsl[lev];
    for (int i = tid; i < nb; i += NT) hist[i] = 0u;
    __syncthreads();
    for (int i = tid; i < NS; i += NT) {
      unsigned k = keys[i];
      if (prefix_bits == 0 || (k >> (32 - prefix_bits)) == prefix)
        atomicAdd(&hist[(k >> shifts[lev]) & (unsigned)(nb - 1)], 1u);
    }
    __syncthreads();
    // inclusive Hillis-Steele scan, ping-pong hist<->scanb
    unsigned* a = hist;
    unsigned* b = scanb;
    for (int off = 1; off < nb; off <<= 1) {
      for (int i = tid; i < nb; i += NT) {
        unsigned v = a[i];
        if (i >= off) v += a[i - off];
        b[i] = v;
      }
      __syncthreads();
      unsigned* t = a; a = b; b = t;
    }
    // locate bin containing rank r
    for (int i = tid; i < nb; i += NT) {
      unsigned c = a[i];
      unsigned p = i ? a[i - 1] : 0u;
      if (r < c && r >= p) { ctrl[0] = (unsigned)i; ctrl[1] = p; }
    }
    __syncthreads();
    unsigned bin = ctrl[0];
    unsigned base = ctrl[1];
    __syncthreads();
    prefix = (prefix << bitsl[lev]) | bin;
    prefix_bits += bitsl[lev];
    r -= base;
  }
  return prefix;  // full 32-bit key
}

// Per-launch prep: zero pooled pad columns; build zero-padded W1 copy [116,16].
__global__ __launch_bounds__(256)
void prep_kernel(const float* __restrict__ W1,
                 float* __restrict__ W1p,
                 float* __restrict__ pooled) {
  int t = blockIdx.x * blockDim.x + threadIdx.x;
  if (t < PSTR * 16) {
    int r = t >> 4;
    W1p[t] = (r < DPOOL) ? W1[t] : 0.f;
  }
  if (t < NBAT * 3) {
    int row = t / 3, p = t - row * 3;
    pooled[(size_t)row * PSTR + DPOOL + p] = 0.f;
  }
}

// One block per (b,f) column: stats + exact quantiles -> pooled[b, :113] (stride 116)
__global__ __launch_bounds__(NT, 1)
void stats_kernel(const float* __restrict__ inputs,
                  const unsigned char* __restrict__ mask,
                  float* __restrict__ pooled) {
  extern __shared__ unsigned char smem[];
  unsigned* keys = (unsigned*)(smem + OFF_KEYS);   // raw float bits, later keys
  unsigned char* mk = smem + OFF_MK;
  unsigned* hist = (unsigned*)(smem + OFF_HIST);
  unsigned* scanb = (unsigned*)(smem + OFF_SCAN);
  unsigned* ctrl = (unsigned*)(smem + OFF_CTRL);
  float* fred = (float*)(smem + OFF_RED);

  const int tid = threadIdx.x;
  const int blk = blockIdx.x;
  const int b = blk >> 4;
  const int f = blk & 15;

  // ---- stage: mask bytes (b128) + strided column gather (b32), async to LDS ----
  {
    const unsigned char* mg = mask + (size_t)b * NS + (size_t)tid * 16;
    async_ld_b128(lds_off32(&mk[tid * 16]), mg);
    const float* ing = inputs + ((size_t)b * NS) * NF + f;
    #pragma unroll
    for (int i = 0; i < NS / NT; ++i) {
      int s = tid + i * NT;
      async_ld_b32(lds_off32(&keys[s]), ing + (size_t)s * NF);
    }
  }
  wait_async0();
  __syncthreads();

  // ---- pass 1: count / sum / min / max ----
  float sum = 0.f, mn = __builtin_inff(), mx = -__builtin_inff(), cntf = 0.f;
  #pragma unroll
  for (int i = 0; i < NS / NT; ++i) {
    int s = tid + i * NT;
    float x = __uint_as_float(keys[s]);
    if (mk[s]) { sum += x; mn = fminf(mn, x); mx = fmaxf(mx, x); cntf += 1.f; }
  }
  cntf = red_op(cntf, fred, tid, 0);
  sum  = red_op(sum,  fred, tid, 0);
  mn   = red_op(mn,   fred, tid, 1);
  mx   = red_op(mx,   fred, tid, 2);
  const int n = (int)(cntf + 0.5f);
  const float safe = fmaxf(cntf, 1.f);
  const float mean = sum / safe;

  // ---- pass 2: exact two-pass variance ----
  float sd = 0.f;
  #pragma unroll
  for (int i = 0; i < NS / NT; ++i) {
    int s = tid + i * NT;
    float x = __uint_as_float(keys[s]);
    if (mk[s]) { float d = x - mean; sd += d * d; }
  }
  sd = red_op(sd, fred, tid, 0);
  const float stdv = sqrtf(sd / fmaxf(safe - 1.f, 1.f));

  // ---- pass 3: convert to order-preserving radix keys (masked-out -> max key) ----
  #pragma unroll
  for (int i = 0; i < NS / NT; ++i) {
    int s = tid + i * NT;
    unsigned u = keys[s];
    unsigned k = (u & 0x80000000u) ? ~u : (u | 0x80000000u);
    keys[s] = mk[s] ? k : 0xFFFFFFFFu;
  }
  __syncthreads();

  // ---- quantiles by exact order-statistic selection ----
  float qv[3] = {0.f, 0.f, 0.f};
  if (n > 0) {
    const float Q[3] = {0.25f, 0.5f, 0.75f};
    for (int qi = 0; qi < 3; ++qi) {
      float pos = Q[qi] * (float)(n - 1);
      int lo = (int)floorf(pos);
      int hi = (int)ceilf(pos);
      lo = lo < 0 ? 0 : (lo > NS - 1 ? NS - 1 : lo);
      hi = hi < 0 ? 0 : (hi > NS - 1 ? NS - 1 : hi);
      float frac = pos - floorf(pos);
      unsigned klo = select_key((unsigned)lo, keys, hist, scanb, ctrl, tid);
      unsigned khi = (hi == lo) ? klo
                                : select_key((unsigned)hi, keys, hist, scanb, ctrl, tid);
      qv[qi] = key_to_float(klo) * (1.f - frac) + key_to_float(khi) * frac;
    }
  }

  if (tid == 0) {
    float* prow = pooled + (size_t)b * PSTR;
    if (n > 0) {
      if (f == 0) prow[0] = cntf;
      prow[1  + f] = mean;
      prow[17 + f] = qv[1];
      prow[33 + f] = mn;
      prow[49 + f] = mx;
      prow[65 + f] = stdv;
      prow[81 + f] = qv[0];
      prow[97 + f] = qv[2];
    } else {
      if (f == 0) prow[0] = 0.f;
      prow[1 + f] = 0.f; prow[17 + f] = 0.f; prow[33 + f] = 0.f;
      prow[49 + f] = 0.f; prow[65 + f] = 0.f; prow[81 + f] = 0.f;
      prow[97 + f] = 0.f;
    }
  }
}

// MLP: pooled[256,116(padded)] -> out[256]. Layer 1 on v_wmma_f32_16x16x4_f32.
// 16 waves, each owns a 16x16 output tile; K = 116 = 29 chunks of 4,
// all fragment loads unconditional (pads are zero in memory).
__global__ __launch_bounds__(512, 1)
void mlp_kernel(const float* __restrict__ pooled,
                const float* __restrict__ W1p,   // [116,16], zero padded
                const float* __restrict__ b1,
                const float* __restrict__ W2, const float* __restrict__ b2,
                const float* __restrict__ W3, const float* __restrict__ b3,
                float* __restrict__ out) {
  __shared__ float h1s[16][16 * 17];  // per-wave 16x16 tile, padded rows

  const int tid = threadIdx.x;
  const int wave = tid >> 5;
  const int lane = tid & 31;
  const int half = lane >> 4;   // 0: K+{0,1} / rows 0..7 of C; 1: K+{2,3} / rows 8..15
  const int l16 = lane & 15;
  const int rowA = (wave << 4) + l16;   // batch row this lane's A fragment covers

  const float* arow = pooled + (size_t)rowA * PSTR + half * 2;  // 8B-aligned
  const float* bcol = W1p + l16 + half * 2 * 16;

  v8f c = {0.f, 0.f, 0.f, 0.f, 0.f, 0.f, 0.f, 0.f};
  #pragma unroll
  for (int kk = 0; kk < PSTR / 4; ++kk) {
    v2f a = *(const v2f*)(arow + kk * 4);
    v2f bf;
    bf.x = bcol[kk * 64 + 0];
    bf.y = bcol[kk * 64 + 16];
    c = __builtin_amdgcn_wmma_f32_16x16x4_f32(false, a, false, bf,
                                              (short)0, c, false, false);
  }

  // scatter C tile (+bias, ReLU) to LDS: vgpr v holds row v (lanes 0-15) / v+8 (16-31)
  const float bias1 = b1[l16];
  #pragma unroll
  for (int v = 0; v < 8; ++v) {
    int row = v + half * 8;
    h1s[wave][row * 17 + l16] = fmaxf(c[v] + bias1, 0.f);
  }
  __syncthreads();

  // layers 2 & 3: one thread per batch row
  if (lane < 16) {
    const float* h1 = &h1s[wave][lane * 17];
    float y = b3[0];
    #pragma unroll
    for (int j = 0; j < 4; ++j) {
      float h2 = b2[j];
      #pragma unroll
      for (int i = 0; i < 16; ++i) h2 += h1[i] * W2[i * 4 + j];
      y += fmaxf(h2, 0.f) * W3[j];
    }
    out[wave * 16 + lane] = y;
  }
}

extern "C" void kernel_launch(void* const* d_in, const int* in_sizes, int n_in,
                              void* d_out, int out_size, void* d_ws, size_t ws_size,
                              hipStream_t stream) {
  const float*         inputs = (const float*)d_in[0];
  const unsigned char* mask   = (const unsigned char*)d_in[1];  // jax bool = 1 byte
  const float* W1 = (const float*)d_in[2];
  const float* b1 = (const float*)d_in[3];
  const float* W2 = (const float*)d_in[4];
  const float* b2 = (const float*)d_in[5];
  const float* W3 = (const float*)d_in[6];
  const float* b3 = (const float*)d_in[7];
  float* out = (float*)d_out;

  float* pooled = (float*)d_ws;                        // [256,116] fp32
  float* W1p    = pooled + (size_t)NBAT * PSTR;        // [116,16] fp32, zero-padded

  prep_kernel<<<dim3(8), dim3(256), 0, stream>>>(W1, W1p, pooled);
  stats_kernel<<<dim3(NBAT * NF), dim3(NT), SMEMB, stream>>>(inputs, mask, pooled);
  mlp_kernel<<<dim3(1), dim3(512), 0, stream>>>(pooled, W1p, b1, W2, b2, W3, b3, out);
}